// UnsignedDotProductPreservingMSELoss_43069932044925
// MI455X (gfx1250) — compile-verified
//
#include <hip/hip_runtime.h>
#include <hip/hip_bf16.h>
#include <cstdint>

// ---------------------------------------------------------------------------
// UnsignedDotProductPreservingMSELoss for MI455X (gfx1250, wave32)
//
// loss = sum_b ((segsum_in[b])^2 - (segsum_tg[b])^2)^2
//      + sum_i (|in[i]| - |tg[i]|)^2
//
// Roofline: 320 MB read @ 23.3 TB/s => ~13.7 us. Memory bound; design goal is
// (a) full-rate streaming, (b) ~50K atomics instead of 20M by exploiting the
// sorted batch array with per-thread run-length accumulation,
// (c) CDNA5 paths: async global->LDS staging + WMMA f32 16x16x4 reduction.
// ---------------------------------------------------------------------------

#define SEG_B   16384      // batch_size from the reference
#define CHUNK   2048       // contiguous elements per thread in the stream pass

typedef float v2f __attribute__((ext_vector_type(2)));
typedef float v8f __attribute__((ext_vector_type(8)));

// ---------------------------------------------------------------------------
// Kernel A: zero the segment accumulators in d_ws and the output scalar.
// (harness poisons d_ws/d_out once and never re-poisons between replays)
// ---------------------------------------------------------------------------
__global__ __launch_bounds__(256) void zero_kernel(float* ws, float* out) {
    int i = blockIdx.x * blockDim.x + threadIdx.x;
    if (i < 2 * SEG_B) ws[i] = 0.0f;
    if (i == 0) out[0] = 0.0f;
}

// ---------------------------------------------------------------------------
// Kernel B: streaming pass. Each thread owns a contiguous CHUNK of elements,
// loads float4 / longlong2 (16B per lane per instr), run-length accumulates
// segment sums (batch is sorted => ~1-2 flushes per thread), and accumulates
// the elementwise |.|-MSE term, wave-reduced before one atomic per wave.
// ---------------------------------------------------------------------------
__device__ inline void flush_seg(float* __restrict__ segIn, float* __restrict__ segTg,
                                 int id, float aIn, float aTg) {
    if (id >= 0) {
        atomicAdd(&segIn[id], aIn);   // global_atomic_add_f32
        atomicAdd(&segTg[id], aTg);
    }
}

__global__ __launch_bounds__(256) void stream_kernel(
        const float* __restrict__ inp, const float* __restrict__ tgt,
        const long long* __restrict__ batch,
        float* __restrict__ segIn, float* __restrict__ segTg,
        float* __restrict__ out, int n) {
    const int tid = blockIdx.x * blockDim.x + threadIdx.x;
    const long long start = (long long)tid * CHUNK;
    float elemAcc = 0.0f;

    if (start < n) {
        long long e = start + CHUNK;
        if (e > n) e = n;                 // N is a multiple of 4
        const int end = (int)e;

        int   cur = -1;
        float aIn = 0.0f, aTg = 0.0f;

        for (int i = (int)start; i < end; i += 4) {
            float4    vi = *(const float4*)(inp + i);
            float4    vt = *(const float4*)(tgt + i);
            longlong2 b0 = *(const longlong2*)(batch + i);
            longlong2 b1 = *(const longlong2*)(batch + i + 2);

            const int   id[4] = {(int)b0.x, (int)b0.y, (int)b1.x, (int)b1.y};
            const float fi[4] = {vi.x, vi.y, vi.z, vi.w};
            const float ft[4] = {vt.x, vt.y, vt.z, vt.w};

#pragma unroll
            for (int j = 0; j < 4; ++j) {
                if (id[j] != cur) {                      // sorted => rare
                    flush_seg(segIn, segTg, cur, aIn, aTg);
                    cur = id[j]; aIn = 0.0f; aTg = 0.0f;
                }
                aIn += fi[j];
                aTg += ft[j];
                float da = __builtin_fabsf(fi[j]) - __builtin_fabsf(ft[j]);
                elemAcc = __builtin_fmaf(da, da, elemAcc);
            }
        }
        flush_seg(segIn, segTg, cur, aIn, aTg);
    }

    // wave32 butterfly reduction of the elementwise loss, 1 atomic per wave
#pragma unroll
    for (int m = 16; m > 0; m >>= 1) elemAcc += __shfl_xor(elemAcc, m, 32);
    if ((threadIdx.x & 31) == 0) atomicAdd(out, elemAcc);
}

// ---------------------------------------------------------------------------
// Kernel C: finalize. One block, 8 waves; each wave owns a 2048-entry slice of
// the B=16384 segment arrays. Slices are staged wave-privately into LDS with
// gfx1250 async loads (ASYNCcnt), then reduced 64-values-at-a-time through
// V_WMMA_F32_16X16X4_F32 with an all-ones B matrix:
//   D[m][n] = sum_k A[m][k] + C[m][n]  -> every column accumulates row sums.
// Lane n holds M=0..7 partials, lane n^16 holds M=8..15 -> one shfl_xor(16).
// ---------------------------------------------------------------------------
__global__ __launch_bounds__(256) void finalize_kernel(
        const float* __restrict__ segIn, const float* __restrict__ segTg,
        float* __restrict__ out) {
    __shared__ float sIn[8][1024];
    __shared__ float sTg[8][1024];

    const int wave = threadIdx.x >> 5;
    const int lane = threadIdx.x & 31;
    const int sliceBase = wave * 2048;          // 8 waves * 2048 = 16384

    v8f acc = {0.f, 0.f, 0.f, 0.f, 0.f, 0.f, 0.f, 0.f};
    const v2f bones = {1.0f, 1.0f};             // all-ones B matrix (4x16)

    for (int t = 0; t < 2; ++t) {
        const int tileBase = sliceBase + t * 1024;

        // Stage 1024 floats of each array, coalesced, async (wave-private).
#pragma unroll
        for (int i = 0; i < 8; ++i) {
            const int off = i * 128 + lane * 4;           // float index
            uint32_t dIn = (uint32_t)(uintptr_t)&sIn[wave][off]; // LDS byte off
            uint32_t dTg = (uint32_t)(uintptr_t)&sTg[wave][off];
            const float* gIn = segIn + tileBase + off;
            const float* gTg = segTg + tileBase + off;
            asm volatile("global_load_async_to_lds_b128 %0, %1, off"
                         :: "v"(dIn), "v"(gIn) : "memory");
            asm volatile("global_load_async_to_lds_b128 %0, %1, off"
                         :: "v"(dTg), "v"(gTg) : "memory");
        }
        asm volatile("s_wait_asynccnt 0" ::: "memory");

        // 16 WMMA steps consume 16*64 = 1024 values.
#pragma unroll
        for (int c = 0; c < 16; ++c) {
            // A-matrix f32 16x4 layout: lanes 0-15 hold K=0,1; lanes 16-31 K=2,3
            const int i0 = c * 64 + ((lane >> 4) * 32) + (lane & 15);
            const int i1 = i0 + 16;
            float si0 = sIn[wave][i0], st0 = sTg[wave][i0];
            float si1 = sIn[wave][i1], st1 = sTg[wave][i1];
            float d0 = si0 * si0 - st0 * st0;
            float d1 = si1 * si1 - st1 * st1;
            v2f a = {d0 * d0, d1 * d1};
            acc = __builtin_amdgcn_wmma_f32_16x16x4_f32(
                      /*neg_a=*/false, a, /*neg_b=*/false, bones,
                      /*c_mod=*/(short)0, acc,
                      /*reuse_a=*/false, /*reuse_b=*/false);
        }
    }

    // Each lane: sum its 8 C/D VGPRs (= its half of the M rows for column n),
    // combine the two M-halves across lane n <-> n^16, one atomic per wave.
    float laneSum = acc[0] + acc[1] + acc[2] + acc[3] +
                    acc[4] + acc[5] + acc[6] + acc[7];
    laneSum += __shfl_xor(laneSum, 16, 32);
    if (lane == 0) atomicAdd(out, laneSum);
}

// ---------------------------------------------------------------------------
extern "C" void kernel_launch(void* const* d_in, const int* in_sizes, int n_in,
                              void* d_out, int out_size, void* d_ws, size_t ws_size,
                              hipStream_t stream) {
    (void)n_in; (void)out_size; (void)ws_size;
    const float*     inp   = (const float*)d_in[0];
    const float*     tgt   = (const float*)d_in[1];
    const long long* batch = (const long long*)d_in[2];
    const int n = in_sizes[0];

    float* segIn = (float*)d_ws;           // [SEG_B]
    float* segTg = segIn + SEG_B;          // [SEG_B]
    float* out   = (float*)d_out;          // scalar

    zero_kernel<<<(2 * SEG_B + 255) / 256, 256, 0, stream>>>(segIn, out);

    const int threads = (n + CHUNK - 1) / CHUNK;
    const int blocks  = (threads + 255) / 256;
    stream_kernel<<<blocks, 256, 0, stream>>>(inp, tgt, batch,
                                              segIn, segTg, out, n);

    finalize_kernel<<<1, 256, 0, stream>>>(segIn, segTg, out);
}